// CodebookForImage_21861383536979
// MI455X (gfx1250) — compile-verified
//
#include <hip/hip_runtime.h>

// Vector quantization: argmin_k ||x_b - c_k||^2 then gather.
// Score = c_sq[k] - 2 * (X @ C^T)[b,k]   (x_sq dropped: constant per row).
// GEMM done with V_WMMA_F32_16X16X4_F32 (exact fp32 -> argmin matches the
// fp32 reference; bf16 WMMA could flip near-ties). 137 GFLOP of matrix work
// vs ~96MB of HBM traffic -> matrix-core bound; codes (32MB) are L2-resident.

#define B_ROWS  8192
#define K_CODES 8192
#define D_EMB   1024
#define NSPLIT  4          // codes range split across blockIdx.y for parallelism

typedef float v2f __attribute__((ext_vector_type(2)));
typedef float v8f __attribute__((ext_vector_type(8)));

// ---------------------------------------------------------------- c_sq[k]
__global__ void __launch_bounds__(256)
csq_kernel(const float* __restrict__ codes, float* __restrict__ csq) {
    const int wave = threadIdx.x >> 5;
    const int lane = threadIdx.x & 31;
    const int row  = blockIdx.x * 8 + wave;
    const float* r = codes + (size_t)row * D_EMB + lane;
    float s = 0.0f;
#pragma unroll
    for (int i = 0; i < D_EMB / 32; ++i) { float v = r[i * 32]; s += v * v; }
#pragma unroll
    for (int off = 16; off > 0; off >>= 1) s += __shfl_xor(s, off, 32);
    if (lane == 0) csq[row] = s;
}

// ------------------------------------------- WMMA cross-term + running argmin
// Wave owns 16 X-rows. A operand (16x4 f32): lane<16 holds X[mBase+lane, k..k+1],
// lane>=16 holds X[mBase+lane-16, k+2..k+3] -> one 8B load per lane per K-step.
// B operand (4x16) mirrors with N per lane -> same addressing from row-major C.
// D/C tile: VGPR v, lane l -> (M = v + 8*(l>=16), N = l&15); running (min,arg)
// kept in that striped layout, reduced across the 16-lane half at the end.
__global__ void __launch_bounds__(256)
argmin_kernel(const float* __restrict__ X, const float* __restrict__ C,
              const float* __restrict__ csq,
              float* __restrict__ partMin, float* __restrict__ partIdx) {
    const int lane  = threadIdx.x & 31;
    const int wave  = threadIdx.x >> 5;
    const int row   = lane & 15;          // M (for A) / N (for B,D) within tile
    const int hi    = lane >> 4;          // K-pair select for A/B operands
    const int mBase = (blockIdx.x * 8 + wave) * 16;
    const int nSplitBase = blockIdx.y * (K_CODES / NSPLIT);

    const float* xrow = X + (size_t)(mBase + row) * D_EMB + hi * 2;

    float runMin[8];
    int   runArg[8];
#pragma unroll
    for (int v = 0; v < 8; ++v) { runMin[v] = 3.0e38f; runArg[v] = 0; }

    for (int g = 0; g < (K_CODES / NSPLIT) / 64; ++g) {
        const int ng = nSplitBase + g * 64;

        v8f acc[4];
#pragma unroll
        for (int t = 0; t < 4; ++t) acc[t] = (v8f){0.f,0.f,0.f,0.f,0.f,0.f,0.f,0.f};

        for (int kc = 0; kc < D_EMB / 64; ++kc) {
            const int k0 = kc * 64;
            // Cache a 16x64 X chunk in 32 VGPRs, reused across 4 code tiles.
            v2f a[16];
#pragma unroll
            for (int s = 0; s < 16; ++s)
                a[s] = *(const v2f*)(xrow + k0 + s * 4);
#pragma unroll
            for (int t = 0; t < 4; ++t) {
                const float* crow =
                    C + (size_t)(ng + t * 16 + row) * D_EMB + hi * 2 + k0;
#pragma unroll
                for (int s = 0; s < 16; ++s) {
                    v2f b = *(const v2f*)(crow + s * 4);
                    acc[t] = __builtin_amdgcn_wmma_f32_16x16x4_f32(
                        false, a[s], false, b, (short)0, acc[t], false, false);
                }
            }
        }

        // Score this group and fold into the running striped (min, argmin).
#pragma unroll
        for (int t = 0; t < 4; ++t) {
            const int   n  = ng + t * 16 + row;   // column for this lane's slots
            const float cs = csq[n];
#pragma unroll
            for (int v = 0; v < 8; ++v) {
                float score = cs - 2.0f * acc[t][v];
                bool  lt    = score < runMin[v];   // strict < keeps first index
                runMin[v] = lt ? score : runMin[v];
                runArg[v] = lt ? n     : runArg[v];
            }
        }
    }

    // Reduce over the 16 N-slots (xor 1,2,4,8 stays within the half-wave).
#pragma unroll
    for (int v = 0; v < 8; ++v) {
        float mv = runMin[v];
        int   ai = runArg[v];
#pragma unroll
        for (int off = 1; off < 16; off <<= 1) {
            float om = __shfl_xor(mv, off, 32);
            int   oa = __shfl_xor(ai, off, 32);
            bool take = (om < mv) || (om == mv && oa < ai);
            mv = take ? om : mv;
            ai = take ? oa : ai;
        }
        if (row == 0) {
            const int m = mBase + v + 8 * hi;
            partMin[m * NSPLIT + blockIdx.y] = mv;
            partIdx[m * NSPLIT + blockIdx.y] = (float)ai;
        }
    }
}

// ------------------------------------------------ merge the NSPLIT candidates
__global__ void __launch_bounds__(256)
merge_kernel(const float* __restrict__ partMin, const float* __restrict__ partIdx,
             float* __restrict__ idxOut) {
    const int row = blockIdx.x * 256 + threadIdx.x;
    float bm = partMin[row * NSPLIT];
    float bi = partIdx[row * NSPLIT];
#pragma unroll
    for (int s = 1; s < NSPLIT; ++s) {
        float m = partMin[row * NSPLIT + s];
        float i = partIdx[row * NSPLIT + s];
        bool take = (m < bm) || (m == bm && i < bi);   // lowest index on ties
        bm = take ? m : bm;
        bi = take ? i : bi;
    }
    idxOut[row] = bi;
}

// ------------------------------------------------ gather: out = x + (q - x)
__global__ void __launch_bounds__(256)
gather_kernel(const float* __restrict__ X, const float* __restrict__ C,
              const float* __restrict__ idxF, float* __restrict__ out) {
    const int b = blockIdx.x;
    const int c = (int)idxF[b];                 // exact: idx < 8192 in fp32
    const float4* q4 = (const float4*)(C + (size_t)c * D_EMB);
    const float4* x4 = (const float4*)(X + (size_t)b * D_EMB);
    float4*       o4 = (float4*)(out + (size_t)b * D_EMB);
    const int t = threadIdx.x;                  // 256 threads * float4 = 1024
    float4 q = q4[t], x = x4[t], o;
    o.x = x.x + (q.x - x.x);
    o.y = x.y + (q.y - x.y);
    o.z = x.z + (q.z - x.z);
    o.w = x.w + (q.w - x.w);
    o4[t] = o;
}

extern "C" void kernel_launch(void* const* d_in, const int* in_sizes, int n_in,
                              void* d_out, int out_size, void* d_ws, size_t ws_size,
                              hipStream_t stream) {
    (void)in_sizes; (void)n_in; (void)out_size; (void)d_ws; (void)ws_size;
    const float* X = (const float*)d_in[0];   // image_tensor [8192,1024] f32
    const float* C = (const float*)d_in[1];   // codes        [8192,1024] f32

    float* O = (float*)d_out;        // [0..8191]: indexes (as float, exact)
    float* U = O + B_ROWS;           // out region [8192 x 1024] (32MB)

    // Scratch carved from the out region (fully overwritten by gather at the
    // end, so no dependence on ws_size): csq | partMin | partIdx.
    float* csq     = U;                                   // 8192 floats
    float* partMin = U + B_ROWS;                          // 8192*NSPLIT
    float* partIdx = U + B_ROWS + B_ROWS * NSPLIT;        // 8192*NSPLIT

    csq_kernel   <<<K_CODES / 8, 256, 0, stream>>>(C, csq);
    argmin_kernel<<<dim3(B_ROWS / 128, NSPLIT), 256, 0, stream>>>(X, C, csq,
                                                                  partMin, partIdx);
    merge_kernel <<<B_ROWS / 256, 256, 0, stream>>>(partMin, partIdx, O);
    gather_kernel<<<B_ROWS, 256, 0, stream>>>(X, C, O, U);
}